// NormActivation_81381040325497
// MI455X (gfx1250) — compile-verified
//
#include <hip/hip_runtime.h>

// irreps: 128x0e + 64x1o + 32x2e  -> DIM = 128*1 + 64*3 + 32*5 = 480
#define DIM            480
#define QUADS_PER_ROW  (DIM / 4)        // 120 float4 per row
#define ROWS_PER_BLOCK 8
#define BLOCK          256
#define EPS            1e-8f

// ---- CDNA5 async global<->LDS path (guarded: fall back to plain copies) ----
#if defined(__has_builtin)
#if __has_builtin(__builtin_amdgcn_global_load_async_to_lds_b128) && \
    __has_builtin(__builtin_amdgcn_global_store_async_from_lds_b128) && \
    __has_builtin(__builtin_amdgcn_s_wait_asynccnt)
#define USE_ASYNC 1
#endif
#endif
#ifndef USE_ASYNC
#define USE_ASYNC 0
#endif

// 16-byte payload type the async b128 builtins expect: v4i32 pointers
typedef int v4i __attribute__((vector_size(16)));
typedef v4i __attribute__((address_space(1)))* gv4p;   // global (AS1)
typedef v4i __attribute__((address_space(3)))* lv4p;   // LDS    (AS3)

__device__ __forceinline__ float sigmoid_clamped(float n) {
    n = fmaxf(n, EPS);
    return 1.0f / (1.0f + __expf(-n));
}

__global__ __launch_bounds__(BLOCK) void
norm_activation_kernel(const float* __restrict__ in, float* __restrict__ out, int nrows) {
    __shared__ float lds[ROWS_PER_BLOCK * DIM];   // 15360 B

    const int tid  = threadIdx.x;
    const int row0 = blockIdx.x * ROWS_PER_BLOCK;
    const int rows = min(ROWS_PER_BLOCK, nrows - row0);
    const int nquad = rows * QUADS_PER_ROW;

    // ---------- Pass 1: global -> LDS, 16B per lane, fully coalesced ----------
    for (int i = tid; i < nquad; i += BLOCK) {
        const int r = i / QUADS_PER_ROW;
        const int q = i - r * QUADS_PER_ROW;
        const long long g = ((long long)(row0 + r)) * DIM + (long long)q * 4;
        const int l = r * DIM + q * 4;
#if USE_ASYNC
        __builtin_amdgcn_global_load_async_to_lds_b128(
            (gv4p)(in + g), (lv4p)&lds[l], 0, 0);
#else
        *(float4*)&lds[l] = *(const float4*)(in + g);
#endif
    }
#if USE_ASYNC
    __builtin_amdgcn_s_wait_asynccnt(0);
#endif
    __syncthreads();

    // ---------- Pass 2: per-irrep scale, in place in LDS ----------
    // (a) 128 scalars (d=1): scale = sigmoid(|x|)
    for (int j = tid; j < rows * 128; j += BLOCK) {
        const int r = j >> 7, c = j & 127;
        const int a = r * DIM + c;
        const float x = lds[a];
        lds[a] = x * sigmoid_clamped(fabsf(x));
    }
    // (b) 64 vectors (d=3), LDS stride 3 (coprime with 64 banks -> conflict free)
    for (int j = tid; j < rows * 64; j += BLOCK) {
        const int r = j >> 6, c = j & 63;
        const int a = r * DIM + 128 + c * 3;
        const float x0 = lds[a], x1 = lds[a + 1], x2 = lds[a + 2];
        const float s = sigmoid_clamped(sqrtf(x0 * x0 + x1 * x1 + x2 * x2));
        lds[a] = x0 * s; lds[a + 1] = x1 * s; lds[a + 2] = x2 * s;
    }
    // (c) 32 irreps (d=5), LDS stride 5 (coprime with 64 banks -> conflict free)
    for (int j = tid; j < rows * 32; j += BLOCK) {
        const int r = j >> 5, c = j & 31;
        const int a = r * DIM + 320 + c * 5;
        const float x0 = lds[a], x1 = lds[a + 1], x2 = lds[a + 2],
                    x3 = lds[a + 3], x4 = lds[a + 4];
        const float s = sigmoid_clamped(
            sqrtf(x0 * x0 + x1 * x1 + x2 * x2 + x3 * x3 + x4 * x4));
        lds[a] = x0 * s; lds[a + 1] = x1 * s; lds[a + 2] = x2 * s;
        lds[a + 3] = x3 * s; lds[a + 4] = x4 * s;
    }
    __syncthreads();

    // ---------- Pass 3: LDS -> global, 16B per lane, fully coalesced ----------
    for (int i = tid; i < nquad; i += BLOCK) {
        const int r = i / QUADS_PER_ROW;
        const int q = i - r * QUADS_PER_ROW;
        const long long g = ((long long)(row0 + r)) * DIM + (long long)q * 4;
        const int l = r * DIM + q * 4;
#if USE_ASYNC
        __builtin_amdgcn_global_store_async_from_lds_b128(
            (gv4p)(out + g), (lv4p)&lds[l], 0, 0);
#else
        *(float4*)(out + g) = *(const float4*)&lds[l];
#endif
    }
#if USE_ASYNC
    __builtin_amdgcn_s_wait_asynccnt(0);  // (also implicit at s_endpgm)
#endif
}

extern "C" void kernel_launch(void* const* d_in, const int* in_sizes, int n_in,
                              void* d_out, int out_size, void* d_ws, size_t ws_size,
                              hipStream_t stream) {
    const float* in = (const float*)d_in[0];
    float* out = (float*)d_out;
    const int nrows = in_sizes[0] / DIM;               // 100000
    const int grid = (nrows + ROWS_PER_BLOCK - 1) / ROWS_PER_BLOCK;  // 12500
    norm_activation_kernel<<<grid, BLOCK, 0, stream>>>(in, out, nrows);
}